// Seq2Seq_33672543600697
// MI455X (gfx1250) — compile-verified
//
#include <hip/hip_runtime.h>
#include <hip/hip_bf16.h>
#include <cstdint>

#define EMB    256
#define HID    512
#define GATES  2048          // 4*HID, PyTorch gate order i,f,g,o
#define VOCAB  32000
#define SRC_LEN 50
#define TRG_LEN 50
#define BATCH  64

#define M_ENC (SRC_LEN * BATCH)          // 3200
#define M_DEC ((TRG_LEN - 1) * BATCH)    // 3136
#define BH    (BATCH * HID)              // 32768

typedef __attribute__((ext_vector_type(2))) float v2f;
typedef __attribute__((ext_vector_type(8))) float v8f;

// ---------------------------------------------------------------------------
// CDNA5 async global->LDS copy (ASYNCcnt path, no VGPR round trip).
// Each lane copies 16B: from the global address in vaddr to the LDS address in vdst. Shared-pointer low 32 bits
// are the LDS byte offset (flat LDS aperture truncates to addr[31:0]).
// ---------------------------------------------------------------------------
__device__ __forceinline__ void async_b128(void* lds, const void* gaddr) {
  unsigned l = (unsigned)(uintptr_t)lds;
  unsigned long long g = (unsigned long long)gaddr;
  asm volatile("global_load_async_to_lds_b128 %0, %1, off" :: "v"(l), "v"(g) : "memory");
}
__device__ __forceinline__ void wait_async0() {
  asm volatile("s_wait_asynccnt 0x0" ::: "memory");
}

// ---------------------------------------------------------------------------
// init: zero the h0/c0 zero-state buffer and grid-barrier words each launch
// ---------------------------------------------------------------------------
__global__ __launch_bounds__(256) void init_ws_kernel(float* zb, unsigned* bar) {
  int i = blockIdx.x * 256 + threadIdx.x;
  if (i < BH) zb[i] = 0.0f;
  if (i < 64) bar[i] = 0u;
}

// ---------------------------------------------------------------------------
// embedding gather: out[r, :] = table[idx[r], :]   (EMB = 256 floats = 64 f4)
// ---------------------------------------------------------------------------
__global__ __launch_bounds__(256) void gather_rows_kernel(
    const int* __restrict__ idx, const float* __restrict__ table,
    float* __restrict__ out, int nrows) {
  int r = blockIdx.x * 4 + (threadIdx.x >> 6);
  int c = threadIdx.x & 63;
  if (r < nrows) {
    const float4* s = reinterpret_cast<const float4*>(table + (size_t)idx[r] * EMB);
    float4*       d = reinterpret_cast<float4*>(out + (size_t)r * EMB);
    d[c] = s[c];
  }
}

// ---------------------------------------------------------------------------
// fp32 WMMA GEMM with bias:  out[M,N] = X[M,K] @ W[N,K]^T + bias[N]
// Block = 256 thr = 8 waves; block tile 64(M) x 256(N); wave tile 32x64.
// K streamed in chunks of 32 through double-buffered LDS filled with
// global_load_async_to_lds_b128; WMMAs consume LDS only.
// Requires M%64==0, N%256==0, K%32==0 (3200/3136, 2048/32000, 256/512).
// ---------------------------------------------------------------------------
#define KC 32
#define G_STRIDE 36                         // pad: 16B-aligned rows, bank-free
#define G_SM_A (64 * G_STRIDE)              // 2304 floats
#define G_SM_B (256 * G_STRIDE)             // 9216 floats
#define G_SM_BUF (G_SM_A + G_SM_B)          // 11520 floats
#define GEMM_SMEM_BYTES (2 * G_SM_BUF * 4)  // 92160 B

__global__ __launch_bounds__(256) void gemm_bias_wmma_f32(
    const float* __restrict__ X, const float* __restrict__ W,
    const float* __restrict__ bias, float* __restrict__ out, int N, int K) {
  extern __shared__ float sg[];

  const int tid  = threadIdx.x;
  const int lane = tid & 31, wid = tid >> 5;
  const int mt   = wid & 1;          // rows mt*32 within block tile
  const int ng   = wid >> 1;         // cols ng*64
  const int r    = lane & 15;
  const int half = lane >> 4;
  const int row0 = blockIdx.y * 64;
  const int col0 = blockIdx.x * 256;

  auto stage = [&](int b, int kc) {
    float* Ab = sg + b * G_SM_BUF;
    float* Bb = sg + b * G_SM_BUF + G_SM_A;
#pragma unroll
    for (int i = 0; i < 2; ++i) {          // A: 64 rows x 8 f4
      int idx = tid + (i << 8), rw = idx >> 3, f4 = idx & 7;
      async_b128(&Ab[rw * G_STRIDE + f4 * 4],
                 X + (size_t)(row0 + rw) * K + kc + f4 * 4);
    }
#pragma unroll
    for (int i = 0; i < 8; ++i) {          // B: 256 rows x 8 f4
      int idx = tid + (i << 8), rw = idx >> 3, f4 = idx & 7;
      async_b128(&Bb[rw * G_STRIDE + f4 * 4],
                 W + (size_t)(col0 + rw) * K + kc + f4 * 4);
    }
  };

  v8f acc[2][4] = {};
  stage(0, 0);
  wait_async0();
  __syncthreads();
  int buf = 0;
  for (int kc = 0; kc < K; kc += KC) {
    if (kc + KC < K) stage(buf ^ 1, kc + KC);
    const float* Ab = sg + buf * G_SM_BUF;
    const float* Bb = sg + buf * G_SM_BUF + G_SM_A;
#pragma unroll
    for (int kk = 0; kk < KC; kk += 4) {
      v2f a0 = *reinterpret_cast<const v2f*>(&Ab[(mt * 32      + r) * G_STRIDE + kk + half * 2]);
      v2f a1 = *reinterpret_cast<const v2f*>(&Ab[(mt * 32 + 16 + r) * G_STRIDE + kk + half * 2]);
#pragma unroll
      for (int nt = 0; nt < 4; ++nt) {
        v2f b = *reinterpret_cast<const v2f*>(&Bb[(ng * 64 + nt * 16 + r) * G_STRIDE + kk + half * 2]);
        acc[0][nt] = __builtin_amdgcn_wmma_f32_16x16x4_f32(false, a0, false, b, (short)0, acc[0][nt], false, false);
        acc[1][nt] = __builtin_amdgcn_wmma_f32_16x16x4_f32(false, a1, false, b, (short)0, acc[1][nt], false, false);
      }
    }
    wait_async0();
    __syncthreads();
    buf ^= 1;
  }

#pragma unroll
  for (int s = 0; s < 2; ++s) {
    const size_t rbase = (size_t)(row0 + mt * 32 + s * 16 + half * 8);
#pragma unroll
    for (int nt = 0; nt < 4; ++nt) {
      int   col = col0 + ng * 64 + nt * 16 + r;
      float bv  = bias[col];
      float* op = out + rbase * (size_t)N + col;
#pragma unroll
      for (int rr = 0; rr < 8; ++rr) op[(size_t)rr * N] = acc[s][nt][rr] + bv;
    }
  }
}

// ---------------------------------------------------------------------------
// Persistent LSTM recurrence. 16 WGs; WG k owns hidden units [32k,32k+32)
// => 128 gate cols {q*512 + 32k + j}. LDS: Whh slice 128x512 (stride 516,
// 258KB) resident for all steps; hprev streamed per step in 16 double-
// buffered async chunks (64x32, stride 36); gate buffer 64x128; cell 64x32.
// Total 323,584 B <= 320KB WGP LDS. Steps separated by L2 grid barrier.
// ---------------------------------------------------------------------------
#define NWG 16
#define WS_STRIDE 516
#define HS_STRIDE 36
#define SM_WHH (128 * WS_STRIDE)   // 66048 floats
#define SM_H   (64 * HS_STRIDE)    //  2304 floats (x2 buffers)
#define SM_G   (64 * 128)          //  8192 floats
#define SM_C   (64 * 32)           //  2048 floats
#define LSTM_SMEM_BYTES ((SM_WHH + 2 * SM_H + SM_G + SM_C) * 4)  // 323584

__device__ __forceinline__ float sigm(float x) { return 1.0f / (1.0f + __expf(-x)); }

__device__ __forceinline__ void grid_barrier(unsigned* cnt, unsigned* gen) {
  __syncthreads();
  if (threadIdx.x == 0) {
    __threadfence();
    unsigned g    = __hip_atomic_load(gen, __ATOMIC_RELAXED, __HIP_MEMORY_SCOPE_AGENT);
    unsigned prev = __hip_atomic_fetch_add(cnt, 1u, __ATOMIC_ACQ_REL, __HIP_MEMORY_SCOPE_AGENT);
    if (prev == NWG - 1u) {
      __hip_atomic_store(cnt, 0u, __ATOMIC_RELAXED, __HIP_MEMORY_SCOPE_AGENT);
      __hip_atomic_store(gen, g + 1u, __ATOMIC_RELEASE, __HIP_MEMORY_SCOPE_AGENT);
    } else {
      while (__hip_atomic_load(gen, __ATOMIC_ACQUIRE, __HIP_MEMORY_SCOPE_AGENT) == g)
        __builtin_amdgcn_s_sleep(2);
    }
  }
  __syncthreads();
}

__global__ __launch_bounds__(256) void lstm_rec_kernel(
    const float* __restrict__ pre,   // [T,B,4H]  (x@Wih^T + b, precomputed)
    const float* __restrict__ Whh,   // [4H,H]
    const float* __restrict__ h0,    // [B,H]
    const float* __restrict__ c0,    // [B,H]
    float* __restrict__ ys,          // [T,B,H]
    float* __restrict__ cT,          // [B,H] or nullptr
    unsigned* __restrict__ bar, int T) {
  extern __shared__ float smem[];
  float* Ws = smem;                          // [128][516] Whh slice
  float* Gs = smem + SM_WHH + 2 * SM_H;      // [64][128] gate pre-activations
  float* Cs = Gs + SM_G;                     // [64][32]  cell state slice
  // hprev chunk buffers live at smem + SM_WHH + buf*SM_H (buf = 0/1)

  const int wg   = blockIdx.x;
  const int tid  = threadIdx.x;
  const int lane = tid & 31;
  const int wid  = tid >> 5;
  const int mt   = wid & 3;    // 4 M-tiles (64 batch rows)
  const int ng   = wid >> 2;   // 2 N-groups of 64 gate cols
  const int r    = lane & 15;
  const int half = lane >> 4;

  // Stage Whh slice once: local gate col n -> global row (n>>5)*512 + wg*32 + (n&31)
  for (int i = tid; i < 128 * 128; i += 256) {
    int n  = i >> 7;
    int kk = (i & 127) << 2;
    int gc = ((n >> 5) << 9) + (wg << 5) + (n & 31);
    float4 w4 = *reinterpret_cast<const float4*>(Whh + (size_t)gc * HID + kk);
    *reinterpret_cast<float4*>(&Ws[n * WS_STRIDE + kk]) = w4;
  }
  for (int i = tid; i < 2048; i += 256) {
    int b = i >> 5, j = i & 31;
    Cs[i] = c0[(size_t)b * HID + (wg << 5) + j];
  }
  __syncthreads();

  for (int t = 0; t < T; ++t) {
    const float* hprev = (t == 0) ? h0 : (ys + (size_t)(t - 1) * BH);
    const float* prow  = pre + (size_t)t * BATCH * GATES;

    auto stage_h = [&](int b, int c) {   // one 64x32 chunk of hprev
      float* Hb = smem + SM_WHH + b * SM_H;
#pragma unroll
      for (int i = 0; i < 2; ++i) {
        int idx = tid + (i << 8), rw = idx >> 3, f4 = idx & 7;
        async_b128(&Hb[rw * HS_STRIDE + f4 * 4],
                   hprev + (size_t)rw * HID + (c << 5) + f4 * 4);
      }
    };

    // G = hprev @ WhhSlice^T  (M=64, N=128, K=512), 16 async K-chunks
    v8f acc[4] = {};
    stage_h(0, 0);
    wait_async0();
    __syncthreads();
    int buf = 0;
    for (int c = 0; c < 16; ++c) {
      if (c + 1 < 16) stage_h(buf ^ 1, c + 1);
      const float* Hb = smem + SM_WHH + buf * SM_H;
#pragma unroll
      for (int kk = 0; kk < 32; kk += 4) {
        v2f a = *reinterpret_cast<const v2f*>(&Hb[(mt * 16 + r) * HS_STRIDE + kk + half * 2]);
#pragma unroll
        for (int nt = 0; nt < 4; ++nt) {
          v2f b = *reinterpret_cast<const v2f*>(
              &Ws[(ng * 64 + nt * 16 + r) * WS_STRIDE + (c << 5) + kk + half * 2]);
          acc[nt] = __builtin_amdgcn_wmma_f32_16x16x4_f32(false, a, false, b, (short)0, acc[nt], false, false);
        }
      }
      wait_async0();
      __syncthreads();
      buf ^= 1;
    }
#pragma unroll
    for (int nt = 0; nt < 4; ++nt) {
      float* gp = &Gs[(mt * 16 + half * 8) * 128 + ng * 64 + nt * 16 + r];
#pragma unroll
      for (int rr = 0; rr < 8; ++rr) gp[rr * 128] = acc[nt][rr];
    }
    __syncthreads();

    // Gates + state update; local col layout: n = q*32 + j, q in {i,f,g,o}
    for (int i = tid; i < 2048; i += 256) {
      int b = i >> 5, j = i & 31;
      const float* pb = prow + (size_t)b * GATES + (wg << 5) + j;
      float gi = Gs[b * 128 +      j] + pb[0];
      float gf = Gs[b * 128 + 32 + j] + pb[512];
      float gg = Gs[b * 128 + 64 + j] + pb[1024];
      float go = Gs[b * 128 + 96 + j] + pb[1536];
      float c  = sigm(gf) * Cs[i] + sigm(gi) * tanhf(gg);
      float h  = sigm(go) * tanhf(c);
      Cs[i] = c;
      ys[(size_t)t * BH + (size_t)b * HID + (wg << 5) + j] = h;
    }
    if (t + 1 < T) grid_barrier(bar, bar + 1);
  }

  if (cT) {  // threads re-read only their own Cs entries
    for (int i = tid; i < 2048; i += 256) {
      int b = i >> 5, j = i & 31;
      cT[(size_t)b * HID + (wg << 5) + j] = Cs[i];
    }
  }
}

// ---------------------------------------------------------------------------
extern "C" void kernel_launch(void* const* d_in, const int* in_sizes, int n_in,
                              void* d_out, int out_size, void* d_ws, size_t ws_size,
                              hipStream_t stream) {
  (void)in_sizes; (void)n_in; (void)out_size; (void)ws_size;
  const int*   src      = (const int*)  d_in[0];
  const int*   trg      = (const int*)  d_in[1];
  const float* enc_emb  = (const float*)d_in[2];
  const float* enc_Wih0 = (const float*)d_in[3];
  const float* enc_Whh0 = (const float*)d_in[4];
  const float* enc_b0   = (const float*)d_in[5];
  const float* enc_Wih1 = (const float*)d_in[6];
  const float* enc_Whh1 = (const float*)d_in[7];
  const float* enc_b1   = (const float*)d_in[8];
  const float* dec_emb  = (const float*)d_in[9];
  const float* dec_Wih0 = (const float*)d_in[10];
  const float* dec_Whh0 = (const float*)d_in[11];
  const float* dec_b0   = (const float*)d_in[12];
  const float* dec_Wih1 = (const float*)d_in[13];
  const float* dec_Whh1 = (const float*)d_in[14];
  const float* dec_b1   = (const float*)d_in[15];
  const float* out_W    = (const float*)d_in[16];
  const float* out_b    = (const float*)d_in[17];
  float* preds = (float*)d_out;

  float* ws = (float*)d_ws;
  size_t o = 0;
  float*    zb  = ws + o; o += BH;                   // zero h0/c0
  unsigned* bar = (unsigned*)(ws + o); o += 64;      // grid barrier words
  float*    c0T = ws + o; o += BH;                   // enc layer0 final c
  float*    c1T = ws + o; o += BH;                   // enc layer1 final c
  float*    xe  = ws + o; o += (size_t)M_ENC * EMB;
  float*    xd  = ws + o; o += (size_t)M_DEC * EMB;
  float*    pre = ws + o; o += (size_t)M_ENC * GATES; // reused by all 4 pre-GEMMs
  float*    y0  = ws + o; o += (size_t)M_ENC * HID;
  float*    y1  = ws + o; o += (size_t)M_ENC * HID;
  float*    d0  = ws + o; o += (size_t)M_DEC * HID;
  float*    d1  = ws + o; o += (size_t)M_DEC * HID;

  (void)hipFuncSetAttribute((const void*)lstm_rec_kernel,
                            hipFuncAttributeMaxDynamicSharedMemorySize, LSTM_SMEM_BYTES);
  (void)hipFuncSetAttribute((const void*)gemm_bias_wmma_f32,
                            hipFuncAttributeMaxDynamicSharedMemorySize, GEMM_SMEM_BYTES);

  init_ws_kernel<<<dim3(BH / 256), dim3(256), 0, stream>>>(zb, bar);

  // ---- encoder ----
  gather_rows_kernel<<<dim3(M_ENC / 4), dim3(256), 0, stream>>>(src, enc_emb, xe, M_ENC);
  gemm_bias_wmma_f32<<<dim3(GATES / 256, M_ENC / 64), dim3(256), GEMM_SMEM_BYTES, stream>>>(
      xe, enc_Wih0, enc_b0, pre, GATES, EMB);
  lstm_rec_kernel<<<dim3(NWG), dim3(256), LSTM_SMEM_BYTES, stream>>>(
      pre, enc_Whh0, zb, zb, y0, c0T, bar, SRC_LEN);
  gemm_bias_wmma_f32<<<dim3(GATES / 256, M_ENC / 64), dim3(256), GEMM_SMEM_BYTES, stream>>>(
      y0, enc_Wih1, enc_b1, pre, GATES, HID);
  lstm_rec_kernel<<<dim3(NWG), dim3(256), LSTM_SMEM_BYTES, stream>>>(
      pre, enc_Whh1, zb, zb, y1, c1T, bar, SRC_LEN);

  // ---- decoder (initial states = encoder final states) ----
  gather_rows_kernel<<<dim3(M_DEC / 4), dim3(256), 0, stream>>>(trg, dec_emb, xd, M_DEC);
  gemm_bias_wmma_f32<<<dim3(GATES / 256, M_DEC / 64), dim3(256), GEMM_SMEM_BYTES, stream>>>(
      xd, dec_Wih0, dec_b0, pre, GATES, EMB);
  lstm_rec_kernel<<<dim3(NWG), dim3(256), LSTM_SMEM_BYTES, stream>>>(
      pre, dec_Whh0, y0 + (size_t)(SRC_LEN - 1) * BH, c0T, d0, nullptr, bar, TRG_LEN - 1);
  gemm_bias_wmma_f32<<<dim3(GATES / 256, M_DEC / 64), dim3(256), GEMM_SMEM_BYTES, stream>>>(
      d0, dec_Wih1, dec_b1, pre, GATES, HID);
  lstm_rec_kernel<<<dim3(NWG), dim3(256), LSTM_SMEM_BYTES, stream>>>(
      pre, dec_Whh1, y1 + (size_t)(SRC_LEN - 1) * BH, c1T, d1, nullptr, bar, TRG_LEN - 1);

  // ---- vocab projection: preds = d1 @ out_W^T + out_b ----
  gemm_bias_wmma_f32<<<dim3(VOCAB / 256, M_DEC / 64), dim3(256), GEMM_SMEM_BYTES, stream>>>(
      d1, out_W, out_b, preds, VOCAB, HID);
}